// ParticleFlowMerger_34316788695093
// MI455X (gfx1250) — compile-verified
//
#include <hip/hip_runtime.h>
#include <stdint.h>

// ParticleFlowMerger for MI455X (gfx1250): pure HBM-bandwidth row scatter.
//   out[b, 0:2, :]  = lepton[b]
//   out[b, 2, :]    = met[b]
//   out[b, 3+j, :]  = jet[b, j, :]   (j < len[b]),  else zeros
// plus lengths (= len+3) and prefix data_mask, flattened into float d_out.
//
// Roofline: ~562 MB moved => ~24 us at 23.3 TB/s. All row/addr math is kept
// scalar (SALU) via a 2D grid (y = batch) + readfirstlane(wave), and the bulk
// 512B rows ride the CDNA5 async global<->LDS path in SADDR form so the only
// vector operand is one reusable lane*16 offset.

#define BQ    8192
#define LMAXQ 64
#define FQ    128
#define SQ    (LMAXQ + 3)      // 67 rows per sample
#define WPB   8                // waves per 256-thread block
#define RTILE 9                // ceil(67 / 8) row-tiles in grid x

typedef float f32x4 __attribute__((ext_vector_type(4)));

__global__ __launch_bounds__(256)
void pf_copy_kernel(const float* __restrict__ jet,
                    const int*   __restrict__ jlen,
                    const float* __restrict__ lepton,
                    const float* __restrict__ met,
                    float*       __restrict__ out)
{
    __shared__ char lbuf[WPB * 512];          // 4 KB staging (of 320 KB/WGP)

    // Wave-uniform scalars: compiler keeps everything below on the SALU.
    const int wave = __builtin_amdgcn_readfirstlane(threadIdx.x >> 5);
    const int lane = threadIdx.x & 31;
    const int r    = blockIdx.x * WPB + wave;  // row within sample, 0..66
    if (r >= SQ) return;                       // uniform branch
    const int b    = blockIdx.y;               // sample index

    const uint32_t voff = (uint32_t)(lane << 4);           // lane byte offset
    const uint64_t dst_base =
        (uint64_t)(uintptr_t)(out + ((size_t)b * SQ + r) * FQ);

    // Select the 512B source row (all wave-uniform scalar branches).
    uint64_t src_base = 0;
    bool     do_copy  = true;
    if (r < 2) {
        src_base = (uint64_t)(uintptr_t)(lepton + ((size_t)b * 2 + r) * FQ);
    } else if (r == 2) {
        src_base = (uint64_t)(uintptr_t)(met + (size_t)b * FQ);
    } else {
        const int j = r - 3;
        if (j < jlen[b]) {
            src_base = (uint64_t)(uintptr_t)(jet + ((size_t)b * LMAXQ + j) * FQ);
        } else {
            do_copy = false;                   // padded slot -> zeros, no read
        }
    }

    if (do_copy) {
        const uint32_t lds =
            (uint32_t)(uintptr_t)lbuf + (uint32_t)(wave * 512) + voff;
        // global -> LDS (ASYNCcnt++), saddr form: one v-offset + SGPR-pair base
        asm volatile("global_load_async_to_lds_b128 %0, %1, %2"
                     :: "v"(lds), "v"(voff), "s"(src_base) : "memory");
        asm volatile("s_wait_asynccnt 0x0" ::: "memory");
        // LDS -> global; drained by S_ENDPGM's implicit wait-idle
        asm volatile("global_store_async_from_lds_b128 %0, %1, %2"
                     :: "v"(voff), "v"(lds), "s"(dst_base) : "memory");
    } else {
        f32x4 z = (f32x4)0.0f;
        __builtin_nontemporal_store(z, (f32x4*)(uintptr_t)dst_base + lane);
    }
}

__global__ __launch_bounds__(96)
void pf_meta_kernel(const int* __restrict__ jlen, float* __restrict__ out)
{
    const int b = blockIdx.x;                  // one block per sample
    const int s = threadIdx.x;                 // 0..95, valid rows 0..66
    const int len = jlen[b] + 3;

    float* lengths_out = out + (size_t)BQ * SQ * FQ;   // B floats
    float* mask_out    = lengths_out + BQ;             // B*S floats

    if (s == 0) lengths_out[b] = (float)len;
    if (s < SQ) mask_out[(size_t)b * SQ + s] = (s < len) ? 1.0f : 0.0f;
}

extern "C" void kernel_launch(void* const* d_in, const int* in_sizes, int n_in,
                              void* d_out, int out_size, void* d_ws, size_t ws_size,
                              hipStream_t stream)
{
    (void)in_sizes; (void)n_in; (void)out_size; (void)d_ws; (void)ws_size;

    const float* jet    = (const float*)d_in[0];   // (B, LMAX, F) f32
    const int*   jlen   = (const int*)  d_in[1];   // (B,) i32
    // d_in[2] = jet_data_mask: unused (guaranteed prefix mask of jlen)
    const float* lepton = (const float*)d_in[3];   // (B, 2, F) f32
    const float* met    = (const float*)d_in[4];   // (B, F) f32
    float*       out    = (float*)d_out;

    dim3 grid(RTILE, BQ);                          // 9 x 8192 blocks
    pf_copy_kernel<<<grid, 256, 0, stream>>>(jet, jlen, lepton, met, out);

    pf_meta_kernel<<<BQ, 96, 0, stream>>>(jlen, out);
}